// BeliefUpdate_5007931867635
// MI455X (gfx1250) — compile-verified
//
#include <hip/hip_runtime.h>
#include <hip/hip_bf16.h>

#ifndef __has_builtin
#define __has_builtin(x) 0
#endif

#if __has_builtin(__builtin_amdgcn_global_load_async_to_lds_b128) && \
    __has_builtin(__builtin_amdgcn_s_wait_asynccnt)
#define BP_ASYNC_LDS 1
#else
#define BP_ASYNC_LDS 0
#endif

#define BP_B      131072
#define BP_D      256
#define BP_EPS    1e-10f
#define BP_THRESH 1e-4f
#define BP_NITER  10

#if BP_ASYNC_LDS
typedef int bp_v4i __attribute__((ext_vector_type(4)));
typedef __attribute__((address_space(1))) bp_v4i* bp_gptr_v4i;
typedef __attribute__((address_space(3))) bp_v4i* bp_lptr_v4i;
#endif

// ---------- small float4 helpers ----------
__device__ __forceinline__ float4 f4_addc(float4 a, float c) {
  return make_float4(a.x + c, a.y + c, a.z + c, a.w + c);
}
__device__ __forceinline__ float4 f4_mul(float4 a, float4 b) {
  return make_float4(a.x * b.x, a.y * b.y, a.z * b.z, a.w * b.w);
}
__device__ __forceinline__ float4 f4_fma(float4 a, float4 b, float4 c) {
  return make_float4(fmaf(a.x, b.x, c.x), fmaf(a.y, b.y, c.y),
                     fmaf(a.z, b.z, c.z), fmaf(a.w, b.w, c.w));
}
__device__ __forceinline__ float4 f4_scale(float4 a, float s) {
  return make_float4(a.x * s, a.y * s, a.z * s, a.w * s);
}
__device__ __forceinline__ float f4_sum(float4 a) { return (a.x + a.y) + (a.z + a.w); }
__device__ __forceinline__ float f4_amaxdiff(float4 a, float4 b) {
  float d0 = fabsf(a.x - b.x), d1 = fabsf(a.y - b.y);
  float d2 = fabsf(a.z - b.z), d3 = fabsf(a.w - b.w);
  return fmaxf(fmaxf(d0, d1), fmaxf(d2, d3));
}

// ---------- wave32 butterfly reductions ----------
__device__ __forceinline__ float wave_sum32(float v) {
#pragma unroll
  for (int off = 16; off >= 1; off >>= 1) v += __shfl_xor(v, off, 32);
  return v;
}
__device__ __forceinline__ float wave_max32(float v) {
#pragma unroll
  for (int off = 16; off >= 1; off >>= 1) v = fmaxf(v, __shfl_xor(v, off, 32));
  return v;
}

// ctrl layout (uint32): [0..9] per-iteration global max-change (float bits, atomic-max),
//                       [10] tstar
__global__ void bp_init_ctrl_kernel(unsigned* __restrict__ ctrl) {
  if (threadIdx.x < 16) ctrl[threadIdx.x] = 0u;
}

// MODE 0: run full 10-iteration unfrozen trajectory, record per-iteration global
//         max change, write belief-after-10.
// MODE 1: if tstar < 10, re-run tstar iterations and overwrite belief output.
template <int MODE>
__global__ __launch_bounds__(256) void bp_pass_kernel(
    const float* __restrict__ prior, const float* __restrict__ lik,
    const float* __restrict__ msgs, float* __restrict__ out,
    unsigned* __restrict__ ctrl) {
  int niter = BP_NITER;
  if (MODE == 1) {
    niter = (int)ctrl[10];
    if (niter >= BP_NITER) return;  // pass-A result already correct (common case)
  }

  const int wid  = (int)(threadIdx.x >> 5);   // wave id in block (8 waves)
  const int lane = (int)(threadIdx.x & 31);
  const int row  = (int)blockIdx.x * 8 + wid; // one row (D=256) per wave
  const size_t base = (size_t)row * BP_D + (size_t)lane * 8;

  float4 p0, p1, l0, l1, q0, q1, r0, r1;

#if BP_ASYNC_LDS
  // CDNA5 path: asynchronously stage all four input streams into LDS,
  // tracked by ASYNCcnt (per-wave), then read back as float4.
  __shared__ __align__(16) float smem[4][8][BP_D];
  {
    const float* gp0 = prior + base;
    const float* gp1 = lik + base;
    const float* gp2 = msgs + base;
    const float* gp3 = msgs + (size_t)BP_B * BP_D + base;
    const float* gp[4] = {gp0, gp1, gp2, gp3};
#pragma unroll
    for (int a = 0; a < 4; ++a) {
      __builtin_amdgcn_global_load_async_to_lds_b128(
          (bp_gptr_v4i)(gp[a]),
          (bp_lptr_v4i)&smem[a][wid][lane * 8], 0, 0);
      __builtin_amdgcn_global_load_async_to_lds_b128(
          (bp_gptr_v4i)(gp[a] + 4),
          (bp_lptr_v4i)&smem[a][wid][lane * 8 + 4], 0, 0);
    }
    __builtin_amdgcn_s_wait_asynccnt(0);  // wave waits for its own async loads only
  }
  p0 = *(const float4*)&smem[0][wid][lane * 8];
  p1 = *(const float4*)&smem[0][wid][lane * 8 + 4];
  l0 = *(const float4*)&smem[1][wid][lane * 8];
  l1 = *(const float4*)&smem[1][wid][lane * 8 + 4];
  q0 = *(const float4*)&smem[2][wid][lane * 8];
  q1 = *(const float4*)&smem[2][wid][lane * 8 + 4];
  r0 = *(const float4*)&smem[3][wid][lane * 8];
  r1 = *(const float4*)&smem[3][wid][lane * 8 + 4];
#else
  p0 = *(const float4*)(prior + base);
  p1 = *(const float4*)(prior + base + 4);
  l0 = *(const float4*)(lik + base);
  l1 = *(const float4*)(lik + base + 4);
  q0 = *(const float4*)(msgs + base);
  q1 = *(const float4*)(msgs + base + 4);
  {
    const float* m1p = msgs + (size_t)BP_B * BP_D + base;
    r0 = *(const float4*)(m1p);
    r1 = *(const float4*)(m1p + 4);
  }
#endif

  // W = (lik+eps)*(m0+eps)*(m1+eps);  exp(log_lik) without any log/exp.
  float4 w0  = f4_mul(f4_mul(f4_addc(l0, BP_EPS), f4_addc(q0, BP_EPS)), f4_addc(r0, BP_EPS));
  float4 w1  = f4_mul(f4_mul(f4_addc(l1, BP_EPS), f4_addc(q1, BP_EPS)), f4_addc(r1, BP_EPS));
  float4 we0 = f4_scale(w0, BP_EPS);   // eps*W, so (b+eps)*W = fma(b, W, eps*W)
  float4 we1 = f4_scale(w1, BP_EPS);

  float4 b0 = p0, b1 = p1;

  __shared__ unsigned chg_sh[BP_NITER];
  if (MODE == 0) {
    if (threadIdx.x < BP_NITER) chg_sh[threadIdx.x] = 0u;
    __syncthreads();
  }

  for (int t = 0; t < niter; ++t) {
    float4 u0 = f4_fma(b0, w0, we0);
    float4 u1 = f4_fma(b1, w1, we1);
    float s   = wave_sum32(f4_sum(u0) + f4_sum(u1));   // row sum over 256 elems
    float rinv = __builtin_amdgcn_rcpf(s);             // v_rcp_f32, 1 ulp
    float4 nb0 = f4_scale(u0, rinv);
    float4 nb1 = f4_scale(u1, rinv);
    if (MODE == 0) {
      float d = fmaxf(f4_amaxdiff(nb0, b0), f4_amaxdiff(nb1, b1));
      d = wave_max32(d);
      if (lane == 0) atomicMax(&chg_sh[t], __float_as_uint(d));  // non-neg float bits monotone
    }
    b0 = nb0;
    b1 = nb1;
  }

  *(float4*)(out + base)     = b0;
  *(float4*)(out + base + 4) = b1;

  if (MODE == 0) {
    __syncthreads();
    if (threadIdx.x < BP_NITER) atomicMax(&ctrl[threadIdx.x], chg_sh[threadIdx.x]);
  }
}

__global__ void bp_finalize_kernel(unsigned* __restrict__ ctrl,
                                   int* __restrict__ iters_out) {
  int tstar = BP_NITER;
  for (int t = 0; t < BP_NITER; ++t) {
    if (__uint_as_float(ctrl[t]) < BP_THRESH) { tstar = t + 1; break; }
  }
  ctrl[10] = (unsigned)tstar;
  *iters_out = tstar;  // iters = first converged iteration (counted), else 10
}

extern "C" void kernel_launch(void* const* d_in, const int* in_sizes, int n_in,
                              void* d_out, int out_size, void* d_ws, size_t ws_size,
                              hipStream_t stream) {
  const float* prior = (const float*)d_in[0];
  const float* lik   = (const float*)d_in[1];
  const float* msgs  = (const float*)d_in[2];
  float* out_belief  = (float*)d_out;
  int* iters_out     = (int*)((float*)d_out + (size_t)BP_B * BP_D);
  unsigned* ctrl     = (unsigned*)d_ws;   // needs 64 bytes of scratch

  bp_init_ctrl_kernel<<<1, 32, 0, stream>>>(ctrl);

  dim3 grid(BP_B / 8), block(256);  // 1 row per wave32, 8 waves/block
  bp_pass_kernel<0><<<grid, block, 0, stream>>>(prior, lik, msgs, out_belief, ctrl);
  bp_finalize_kernel<<<1, 1, 0, stream>>>(ctrl, iters_out);
  bp_pass_kernel<1><<<grid, block, 0, stream>>>(prior, lik, msgs, out_belief, ctrl);
}